// AUCLoss_1_89962384982112
// MI455X (gfx1250) — compile-verified
//
#include <hip/hip_runtime.h>
#include <hip/hip_bf16.h>

// AUC-style pairwise loss, N=4096, D=128, C=10.
// G = pred@pred.T via V_WMMA_F32_16X16X4_F32, per-class sums collapsed to
// (Sw, Sm=SumwM, Sm2=SumwM^2); loss1 = (Sw - 2Sm + Sm2)/(exist-1), etc.

#define N_ROWS 4096
#define N_DIM  128
#define N_CLS  10
#define EPSF   1e-8f

typedef __attribute__((ext_vector_type(2))) float v2f;
typedef __attribute__((ext_vector_type(8))) float v8f;

// ---------------- workspace layout (floats) ----------------
// [0..29]    global accumulators: c*3 + {Sw, Sm, Sm2}
// [32..41]   int counts[10]
// [48..57]   invcnt[10]
// [64..73]   invwn[10]
// [80..80+4096)          sq[i]
// [80+4096 .. +4096*10)  proj[i*10+c]
#define WS_ACC    0
#define WS_CNT    32
#define WS_ICNT   48
#define WS_IWN    64
#define WS_SQ     80
#define WS_PROJ   (80 + N_ROWS)

__global__ void k_init(float* __restrict__ acc, int* __restrict__ cnt) {
  int t = threadIdx.x;
  if (t < 30) acc[t] = 0.0f;
  if (t < N_CLS) cnt[t] = 0;
}

__global__ void k_counts(const int* __restrict__ tgt, int* __restrict__ cnt) {
  int i = blockIdx.x * 256 + threadIdx.x;
  if (i < N_ROWS) atomicAdd(&cnt[tgt[i]], 1);
}

__global__ void k_scal(const float* __restrict__ W, const int* __restrict__ cnt,
                       float* __restrict__ invwn, float* __restrict__ invcnt) {
  int c = threadIdx.x;
  if (c < N_CLS) {
    float s = 0.0f;
    for (int k = 0; k < N_DIM; ++k) { float w = W[c * N_DIM + k]; s += w * w; }
    float wn = fmaxf(sqrtf(s), EPSF);
    invwn[c] = 1.0f / wn;
    int n = cnt[c];
    invcnt[c] = 1.0f / (float)(n < 1 ? 1 : n);
  }
}

// one wave per row: sq[i] and proj[i][0..9]
__global__ __launch_bounds__(256) void k_rows(const float* __restrict__ pred,
                                              const float* __restrict__ W,
                                              float* __restrict__ sq,
                                              float* __restrict__ proj) {
  const int tid  = threadIdx.x;
  const int lane = tid & 31;
  const int wave = tid >> 5;
  const int row  = blockIdx.x * 8 + wave;           // grid = 512 blocks
  const float4 p = *(const float4*)(pred + row * N_DIM + lane * 4);
  float s = p.x * p.x + p.y * p.y + p.z * p.z + p.w * p.w;
  for (int off = 16; off > 0; off >>= 1) s += __shfl_down(s, off, 32);
  if (lane == 0) sq[row] = s;
#pragma unroll
  for (int c = 0; c < N_CLS; ++c) {
    const float4 w4 = *(const float4*)(W + c * N_DIM + lane * 4);
    float d = p.x * w4.x + p.y * w4.y + p.z * w4.z + p.w * w4.w;
    for (int off = 16; off > 0; off >>= 1) d += __shfl_down(d, off, 32);
    if (lane == 0) proj[row * N_CLS + c] = d;
  }
}

// main pairwise kernel: one 16x16 tile of G per wave via WMMA f32 16x16x4
__global__ __launch_bounds__(256) void k_pairs(const float* __restrict__ pred,
                                               const int* __restrict__ tgt,
                                               const float* __restrict__ sq,
                                               const float* __restrict__ proj,
                                               const float* __restrict__ invcnt,
                                               const float* __restrict__ invwn,
                                               float* __restrict__ gacc) {
  __shared__ float lacc[30];
  const int tid = threadIdx.x;
  if (tid < 30) lacc[tid] = 0.0f;
  __syncthreads();

  const int lane = tid & 31;
  const int wave = tid >> 5;
  const int tile = blockIdx.x * 8 + wave;           // 65536 tiles (256x256)
  const int i0   = (tile >> 8) << 4;
  const int j0   = (tile & 255) << 4;

  const int half = lane >> 4;                       // 0 or 1
  const int l15  = lane & 15;
  const int kb   = half * 2;                        // K-pair base for this lane

  // A(16x4 f32): lane l -> row i0+(l&15), K = kb, kb+1
  // B(4x16 f32): lane l -> col j0+(l&15), K = kb, kb+1  (same addressing)
  const float* pa = pred + (i0 + l15) * N_DIM + kb;
  const float* pb = pred + (j0 + l15) * N_DIM + kb;

  v8f cacc = {};
#pragma unroll
  for (int kk = 0; kk < N_DIM / 4; ++kk) {
    v2f a = *(const v2f*)(pa + 4 * kk);
    v2f b = *(const v2f*)(pb + 4 * kk);
    cacc = __builtin_amdgcn_wmma_f32_16x16x4_f32(
        /*neg_a=*/false, a, /*neg_b=*/false, b,
        /*c_mod=*/(short)0, cacc, /*reuse_a=*/false, /*reuse_b=*/false);
  }

  // C/D layout: lane l, vgpr v -> M = v + 8*(l>>4), Nc = l&15
  const int j    = j0 + l15;
  const float sqj = sq[j];
  const int tj   = tgt[j];
  const float icj = invcnt[tj];

  float accW[N_CLS], accM[N_CLS], accM2[N_CLS];
#pragma unroll
  for (int c = 0; c < N_CLS; ++c) { accW[c] = 0.f; accM[c] = 0.f; accM2[c] = 0.f; }

#pragma unroll
  for (int v = 0; v < 8; ++v) {
    const int i     = i0 + v + half * 8;
    const float g   = cacc[v];
    const float sqi = sq[i];
    const int ti    = tgt[i];
    float d2 = sqi + sqj - 2.0f * g;
    d2 = fmaxf(d2, 0.0f);
    const float dn = fmaxf(sqrtf(d2 + 1e-16f), EPSF);
    const float Mv = (proj[i * N_CLS + ti] - proj[j * N_CLS + ti]) * invwn[ti] / dn;
    const float w  = (ti != tj) ? invcnt[ti] * icj : 0.0f;
#pragma unroll
    for (int c = 0; c < N_CLS; ++c) {
      const float wv = (ti == c) ? w : 0.0f;
      accW[c] += wv;
      const float t = wv * Mv;
      accM[c]  += t;
      accM2[c] += t * Mv;
    }
  }

  // wave reduce, then LDS atomics, then one global atomic per slot per block
#pragma unroll
  for (int c = 0; c < N_CLS; ++c) {
    float a0 = accW[c], a1 = accM[c], a2 = accM2[c];
    for (int off = 16; off > 0; off >>= 1) {
      a0 += __shfl_down(a0, off, 32);
      a1 += __shfl_down(a1, off, 32);
      a2 += __shfl_down(a2, off, 32);
    }
    if (lane == 0) {
      atomicAdd(&lacc[c * 3 + 0], a0);
      atomicAdd(&lacc[c * 3 + 1], a1);
      atomicAdd(&lacc[c * 3 + 2], a2);
    }
  }
  __syncthreads();
  if (tid < 30) unsafeAtomicAdd(&gacc[tid], lacc[tid]);
}

__global__ void k_final(const float* __restrict__ acc, const int* __restrict__ cnt,
                        float* __restrict__ out) {
  if (threadIdx.x == 0) {
    int ex = 0;
    for (int c = 0; c < N_CLS; ++c) ex += (cnt[c] > 0) ? 1 : 0;
    const float exist = (float)ex;
    const float denom = exist - 1.0f;
    float l1 = 0.0f, l2 = 0.0f;
    for (int c = 0; c < N_CLS; ++c) {
      if (cnt[c] > 0) {
        const float Sw = acc[c * 3 + 0], Sm = acc[c * 3 + 1], Sm2 = acc[c * 3 + 2];
        l1 += (Sw - 2.0f * Sm + Sm2) / denom;
        const float mm = Sm / denom;
        const float mv = (Sm2 - 2.0f * mm * Sm + mm * mm * Sw) / denom;
        l2 += fabsf(mv / mm);
      }
    }
    out[0] = l1 / exist;
    out[1] = l2 / exist;
  }
}

extern "C" void kernel_launch(void* const* d_in, const int* in_sizes, int n_in,
                              void* d_out, int out_size, void* d_ws, size_t ws_size,
                              hipStream_t stream) {
  const float* pred = (const float*)d_in[0];   // 4096x128 f32
  const int*   tgt  = (const int*)d_in[1];     // 4096 int
  const float* W    = (const float*)d_in[2];   // 10x128 f32
  float* out = (float*)d_out;                  // (loss1, loss2)

  float* ws     = (float*)d_ws;                // needs 45136 floats (~180 KB)
  float* acc    = ws + WS_ACC;
  int*   cnt    = (int*)(ws + WS_CNT);
  float* invcnt = ws + WS_ICNT;
  float* invwn  = ws + WS_IWN;
  float* sq     = ws + WS_SQ;
  float* proj   = ws + WS_PROJ;

  k_init  <<<1, 64, 0, stream>>>(acc, cnt);
  k_counts<<<N_ROWS / 256, 256, 0, stream>>>(tgt, cnt);
  k_scal  <<<1, 32, 0, stream>>>(W, cnt, invwn, invcnt);
  k_rows  <<<N_ROWS / 8, 256, 0, stream>>>(pred, W, sq, proj);
  k_pairs <<<(N_ROWS / 16) * (N_ROWS / 16) / 8, 256, 0, stream>>>(
      pred, tgt, sq, proj, invcnt, invwn, acc);
  k_final <<<1, 32, 0, stream>>>(acc, cnt, out);
}